// EarthMoversDistance_54631984005442
// MI455X (gfx1250) — compile-verified
//
#include <hip/hip_runtime.h>

// ---------------------------------------------------------------------------
// Entropic EMD (Sinkhorn) for 16 x (2048 x 3) point-cloud pairs, f32.
// Cost tiles recomputed on the fly with V_WMMA_F32_16X16X4_F32:
//   A row m = (-2x,-2y,-2z, 1), B col n = (x,y,z, rb_n) => A.B = rb - 2 a.b
//   accumulator = loop-invariant ra_m (separate VDST/SRC2, no per-tile copies)
//   => D = |a-b|^2
// sqrt guard via free |.| source modifier; all LSE math in base 2 with bare
// v_sqrt_f32 / v_exp_f32 / v_log_f32 and a group-of-4 chunked online max.
// ---------------------------------------------------------------------------

typedef float v2f __attribute__((ext_vector_type(2)));
typedef float v8f __attribute__((ext_vector_type(8)));

#define NPTS        2048
#define NB          16
#define CTILES      128          // NPTS / 16 column tiles
#define CGROUPS     32           // CTILES / 4
#define BLOCK       256          // 8 wave32s
#define WAVES       8
#define ROWS_PER_WG 128          // 8 waves * 16 rows
#define RGRPS       16           // NPTS / ROWS_PER_WG
#define NITERS      200

#define K2          28.8539008178f   // log2(e)/eps = 20 * 1.4426950409
#define EPS_LN2     0.0346573590f    // eps * ln2
#define NEG_EPS_LOGW 0.3812309493f   // -eps * log(1/2048)
#define TW2         (-22.0f)         // 2*log(1/2048)*log2(e) == -22 exactly

// base-2 log-sum-exp merge: state = (max, sum of 2^(x-max))
__device__ __forceinline__ void lse2_merge(float& m, float& s, float om, float os) {
    float nm = fmaxf(m, om);
    s = s * __builtin_amdgcn_exp2f(m - nm) + os * __builtin_amdgcn_exp2f(om - nm);
    m = nm;
}

// One Sinkhorn half-step:
//   dual_out[i] = -eps * ( log_w + LSE_j( (dual_in[j] - C_ij)/eps ) )
__global__ __launch_bounds__(BLOCK)
void sinkhorn_half_kernel(const float* __restrict__ Pcloud,
                          const float* __restrict__ Qcloud,
                          const float* __restrict__ dual_in,
                          float* __restrict__ dual_out)
{
    __shared__ float4 Q4[NPTS];       // column cloud (x, y, z, rb)
    __shared__ float  din[NPTS];      // dual_in * log2(e)/eps
    __shared__ float4 P4[ROWS_PER_WG];// row cloud (-2x,-2y,-2z, ra)

    const int batch = blockIdx.y;
    const int rgrp  = blockIdx.x;
    const int tid   = threadIdx.x;
    const int wave  = tid >> 5;
    const int lane  = tid & 31;
    const int half  = lane >> 4;      // WMMA half-wave (rows v / v+8, K 0,1 / 2,3)
    const int l16   = lane & 15;

    const float* qc = Qcloud + batch * NPTS * 3;
    for (int i = tid; i < NPTS; i += BLOCK) {
        float x = qc[3 * i], y = qc[3 * i + 1], z = qc[3 * i + 2];
        Q4[i] = make_float4(x, y, z, x * x + y * y + z * z);
        din[i] = dual_in[batch * NPTS + i] * K2;
    }
    const float* pcld = Pcloud + (batch * NPTS + rgrp * ROWS_PER_WG) * 3;
    for (int i = tid; i < ROWS_PER_WG; i += BLOCK) {
        float x = pcld[3 * i], y = pcld[3 * i + 1], z = pcld[3 * i + 2];
        P4[i] = make_float4(-2.0f * x, -2.0f * y, -2.0f * z,
                            x * x + y * y + z * z);
    }
    __syncthreads();

    const int base = wave * 16;       // this wave's 16-row tile (local)

    // A fragment (16x4 f32): lanes 0-15 hold K=0,1 ; lanes 16-31 hold K=2,3.
    // K=3 element is 1 (multiplies B's rb row).
    float4 ap = P4[base + l16];
    v2f afrag;
    afrag.x = half ? ap.z : ap.x;     // -2z   | -2x
    afrag.y = half ? 1.0f : ap.y;     //  1    | -2y

    // Loop-invariant accumulator: ra_m per accumulator slot (VDST != SRC2 in
    // VOP3P, so no per-tile register copies are needed).
    v8f cinit;
#pragma unroll
    for (int v = 0; v < 8; ++v) cinit[v] = P4[base + 8 * half + v].w;

    float m[8], s[8];
#pragma unroll
    for (int v = 0; v < 8; ++v) { m[v] = -__builtin_huge_valf(); s[v] = 0.0f; }

    for (int cg = 0; cg < CGROUPS; ++cg) {
        float y4[4][8];
#pragma unroll
        for (int u = 0; u < 4; ++u) {
            const int col = (cg * 4 + u) * 16 + l16;
            float4 bq = Q4[col];
            v2f bfrag;                // B (4x16): K=0,1 | K=2,3 ; K=3 row is rb
            bfrag.x = half ? bq.z : bq.x;
            bfrag.y = half ? bq.w : bq.y;
            const float dinv = din[col];

            // D = ra + (rb - 2 a.b) = |a-b|^2
            v8f d = __builtin_amdgcn_wmma_f32_16x16x4_f32(false, afrag, false, bfrag,
                                                          (short)0, cinit, false, false);
#pragma unroll
            for (int v = 0; v < 8; ++v) {
                // |.| is a free source modifier on v_sqrt_f32 (guards tiny
                // negative cancellation in d2 without maxnum canonicalize).
                float cst = __builtin_amdgcn_sqrtf(__builtin_fabsf(d[v]));
                // base-2 exponent: (g_j - C_ij)/eps * log2(e)
                y4[u][v] = __builtin_fmaf(cst, -K2, dinv);
            }
        }
        // chunked online LSE: one rescale + 4 term-exps per 4 elements
#pragma unroll
        for (int v = 0; v < 8; ++v) {
            float gm = fmaxf(fmaxf(y4[0][v], y4[1][v]), fmaxf(y4[2][v], y4[3][v]));
            float nm = fmaxf(m[v], gm);
            float sr = s[v] * __builtin_amdgcn_exp2f(m[v] - nm);
            sr += __builtin_amdgcn_exp2f(y4[0][v] - nm);
            sr += __builtin_amdgcn_exp2f(y4[1][v] - nm);
            sr += __builtin_amdgcn_exp2f(y4[2][v] - nm);
            sr += __builtin_amdgcn_exp2f(y4[3][v] - nm);
            s[v] = sr; m[v] = nm;
        }
    }

    // Reduce LSE across the 16 lanes sharing each row (wave32, width-16 xor tree)
    float res[8];
#pragma unroll
    for (int v = 0; v < 8; ++v) {
        float mm = m[v], ss = s[v];
#pragma unroll
        for (int off = 1; off < 16; off <<= 1) {
            float om = __shfl_xor(mm, off, 16);
            float os = __shfl_xor(ss, off, 16);
            lse2_merge(mm, ss, om, os);
        }
        // f = -eps*(log_w + ln2*(m + log2 s))
        res[v] = NEG_EPS_LOGW - EPS_LN2 * (mm + __builtin_amdgcn_logf(ss));
    }
    if (l16 == 0) {
#pragma unroll
        for (int v = 0; v < 8; ++v)
            dual_out[batch * NPTS + rgrp * ROWS_PER_WG + base + 8 * half + v] = res[v];
    }
}

// Final contraction: sum_ij exp(2logw + (f_i+g_j)/eps - C_ij/eps) * C_ij
// Plan entries are <= O(1), so direct exp2 (underflow is benign) - 1 exp/element.
__global__ __launch_bounds__(BLOCK)
void transport_sum_kernel(const float* __restrict__ Pcloud,
                          const float* __restrict__ Qcloud,
                          const float* __restrict__ fdual,
                          const float* __restrict__ gdual,
                          float* __restrict__ partials)
{
    __shared__ float4 Q4[NPTS];
    __shared__ float  gin[NPTS];
    __shared__ float4 P4[ROWS_PER_WG];
    __shared__ float  fl[ROWS_PER_WG];
    __shared__ float  red[WAVES];

    const int batch = blockIdx.y;
    const int rgrp  = blockIdx.x;
    const int tid   = threadIdx.x;
    const int wave  = tid >> 5;
    const int lane  = tid & 31;
    const int half  = lane >> 4;
    const int l16   = lane & 15;

    const float* qc = Qcloud + batch * NPTS * 3;
    for (int i = tid; i < NPTS; i += BLOCK) {
        float x = qc[3 * i], y = qc[3 * i + 1], z = qc[3 * i + 2];
        Q4[i] = make_float4(x, y, z, x * x + y * y + z * z);
        gin[i] = gdual[batch * NPTS + i] * K2;
    }
    const float* pcld = Pcloud + (batch * NPTS + rgrp * ROWS_PER_WG) * 3;
    for (int i = tid; i < ROWS_PER_WG; i += BLOCK) {
        float x = pcld[3 * i], y = pcld[3 * i + 1], z = pcld[3 * i + 2];
        P4[i] = make_float4(-2.0f * x, -2.0f * y, -2.0f * z,
                            x * x + y * y + z * z);
        fl[i] = fdual[batch * NPTS + rgrp * ROWS_PER_WG + i] * K2;
    }
    __syncthreads();

    const int base = wave * 16;
    float4 ap = P4[base + l16];
    v2f afrag;
    afrag.x = half ? ap.z : ap.x;
    afrag.y = half ? 1.0f : ap.y;

    v8f cinit;
    float fver[8];
#pragma unroll
    for (int v = 0; v < 8; ++v) {
        cinit[v] = P4[base + 8 * half + v].w;
        fver[v]  = fl[base + 8 * half + v];
    }

    float acc = 0.0f;
    for (int ct = 0; ct < CTILES; ++ct) {
        const int col = ct * 16 + l16;
        float4 bq = Q4[col];
        v2f bfrag;
        bfrag.x = half ? bq.z : bq.x;
        bfrag.y = half ? bq.w : bq.y;
        const float gj = gin[col];

        v8f d = __builtin_amdgcn_wmma_f32_16x16x4_f32(false, afrag, false, bfrag,
                                                      (short)0, cinit, false, false);
#pragma unroll
        for (int v = 0; v < 8; ++v) {
            float cst = __builtin_amdgcn_sqrtf(__builtin_fabsf(d[v]));
            float lp2 = TW2 + fver[v] + gj - cst * K2;   // base-2 log plan entry
            acc = __builtin_fmaf(__builtin_amdgcn_exp2f(lp2), cst, acc);
        }
    }

    // deterministic block reduction
#pragma unroll
    for (int off = 1; off < 32; off <<= 1) acc += __shfl_xor(acc, off, 32);
    if (lane == 0) red[wave] = acc;
    __syncthreads();
    if (tid == 0) {
        float ssum = 0.0f;
        for (int w = 0; w < WAVES; ++w) ssum += red[w];
        partials[batch * RGRPS + rgrp] = ssum;
    }
}

__global__ void zero_kernel(float* __restrict__ p, int n) {
    int i = blockIdx.x * blockDim.x + threadIdx.x;
    if (i < n) p[i] = 0.0f;
}

__global__ void reduce_kernel(const float* __restrict__ partials,
                              float* __restrict__ out) {
    if (threadIdx.x == 0) {
        float ssum = 0.0f;
        for (int i = 0; i < NB * RGRPS; ++i) ssum += partials[i];
        out[0] = ssum;
    }
}

extern "C" void kernel_launch(void* const* d_in, const int* in_sizes, int n_in,
                              void* d_out, int out_size, void* d_ws, size_t ws_size,
                              hipStream_t stream) {
    const float* pc1 = (const float*)d_in[0];
    const float* pc2 = (const float*)d_in[1];

    float* f        = (float*)d_ws;            // NB*NPTS
    float* g        = f + NB * NPTS;           // NB*NPTS
    float* partials = g + NB * NPTS;           // NB*RGRPS

    const int ztot = 2 * NB * NPTS;
    zero_kernel<<<(ztot + 255) / 256, 256, 0, stream>>>(f, ztot);

    dim3 grid(RGRPS, NB);
    for (int it = 0; it < NITERS; ++it) {
        sinkhorn_half_kernel<<<grid, BLOCK, 0, stream>>>(pc1, pc2, g, f);
        sinkhorn_half_kernel<<<grid, BLOCK, 0, stream>>>(pc2, pc1, f, g);
    }

    transport_sum_kernel<<<grid, BLOCK, 0, stream>>>(pc1, pc2, f, g, partials);
    reduce_kernel<<<1, 32, 0, stream>>>(partials, (float*)d_out);
}